// LinkPredictorModel_7834020348027
// MI455X (gfx1250) — compile-verified
//
#include <hip/hip_runtime.h>
#include <hip/hip_bf16.h>

typedef __attribute__((ext_vector_type(2))) float v2f;
typedef __attribute__((ext_vector_type(8))) float v8f;

#define HID 64

// ---------------- degree / norm ----------------

__global__ void k_init_deg(int* deg, int n) {
    int i = blockIdx.x * blockDim.x + threadIdx.x;
    if (i < n) deg[i] = 1;  // self loop
}

__global__ void k_count_deg(const int* __restrict__ ei, int* deg, int E) {
    int e = blockIdx.x * blockDim.x + threadIdx.x;
    if (e < E) atomicAdd(&deg[ei[E + e]], 1);  // dst-indegree
}

__global__ void k_dinv(const int* __restrict__ deg, float* dinv, int n) {
    int i = blockIdx.x * blockDim.x + threadIdx.x;
    if (i < n) dinv[i] = rsqrtf((float)deg[i]);
}

__global__ void k_norm(const int* __restrict__ ei, const float* __restrict__ dinv,
                       float* nrm, int E) {
    int e = blockIdx.x * blockDim.x + threadIdx.x;
    if (e < E) nrm[e] = dinv[ei[e]] * dinv[ei[E + e]];
}

// ---------------- WMMA GEMM: Y[n,64] = X[n,K] @ W[K,64] (+bias)(+relu) ----------------
// One wave per 16-row slab; 4 output tiles of 16x16; K chained in chunks of 4 via
// V_WMMA_F32_16X16X4_F32 (fp32-exact).
// A layout (16x4 f32): lanes 0-15 rows M=0..15, VGPR0/1 = K0/K1; lanes 16-31 = K2/K3.
// C/D layout: VGPR r: lanes 0-15 -> (M=r, N=lane), lanes 16-31 -> (M=r+8, N=lane-16).

__global__ void k_gemm64(const float* __restrict__ X, const float* __restrict__ W,
                         const float* __restrict__ bias, float* __restrict__ Y,
                         int n, int K, int relu) {
    int lane = threadIdx.x;                       // 0..31
    int rowBase = (blockIdx.x * blockDim.y + threadIdx.y) * 16;
    if (rowBase >= n) return;                     // wave-uniform guard (EXEC stays all-1)
    int hl = lane & 15;
    int kh = lane >> 4;                           // K-pair select
    int kChunks = K >> 2;

    for (int t = 0; t < 4; ++t) {                 // output col tiles
        v8f acc = {0.f, 0.f, 0.f, 0.f, 0.f, 0.f, 0.f, 0.f};
        for (int kk = 0; kk < kChunks; ++kk) {
            int k0 = kk * 4 + kh * 2;
            const float* ap = X + (size_t)(rowBase + hl) * K + k0;
            v2f a; a.x = ap[0]; a.y = ap[1];
            const float* bp = W + (size_t)k0 * HID + t * 16 + hl;
            v2f b; b.x = bp[0]; b.y = bp[HID];
            acc = __builtin_amdgcn_wmma_f32_16x16x4_f32(
                false, a, false, b, (short)0, acc, false, false);
        }
        int col = t * 16 + hl;
        float bv = bias ? bias[col] : 0.0f;
        int row0 = rowBase + kh * 8;
#pragma unroll
        for (int r = 0; r < 8; ++r) {
            float v = acc[r] + bv;
            if (relu) v = fmaxf(v, 0.0f);
            Y[(size_t)(row0 + r) * HID + col] = v;
        }
    }
}

// ---------------- aggregation ----------------

// tmp[n,:] = hw[n,:] * dinv[n]^2  (self-loop term; also zero-initializes tmp)
__global__ void k_selfinit(const float* __restrict__ hw, const float* __restrict__ dinv,
                           float* tmp, int n) {
    int t = blockIdx.x * blockDim.x + threadIdx.x;
    if (t >= n * HID) return;
    int node = t >> 6;
    float dv = dinv[node];
    tmp[t] = hw[t] * dv * dv;
}

// 16 threads per edge, float4 each: tmp[dst,:] += norm[e] * hw[src,:]
// Lanes 16t..16t+15 cover one contiguous 256B row -> coalesced b128 loads + atomics.
__global__ void k_scatter(const int* __restrict__ ei, const float* __restrict__ nrm,
                          const float* __restrict__ hw, float* tmp, int E) {
    int t = blockIdx.x * blockDim.x + threadIdx.x;
    if (t >= E * 16) return;
    int e = t >> 4;
    int g = t & 15;
    int s = ei[e];
    int d = ei[E + e];
    float nv = nrm[e];
    float4 m = ((const float4*)(hw + (size_t)s * HID))[g];
    float* o = tmp + (size_t)d * HID + g * 4;
    atomicAdd(o + 0, m.x * nv);
    atomicAdd(o + 1, m.y * nv);
    atomicAdd(o + 2, m.z * nv);
    atomicAdd(o + 3, m.w * nv);
}

// h = relu(tmp + b)  in place
__global__ void k_bias_relu(float* tmp, const float* __restrict__ b, int n) {
    int t = blockIdx.x * blockDim.x + threadIdx.x;
    if (t >= n * HID) return;
    tmp[t] = fmaxf(tmp[t] + b[t & (HID - 1)], 0.0f);
}

// ---------------- edge scoring ----------------
// out[e] = relu(u[src] + v[dst]) . lpW2 + lpb2   (lpb1 already folded into u)
// 16 lanes per edge: each lane handles one float4 chunk of the 64-wide row, so the
// half-wave issues one coalesced 256B gather per operand; butterfly-reduce over 16 lanes.

__global__ void k_edge_score(const int* __restrict__ ei, const float* __restrict__ u,
                             const float* __restrict__ v, const float* __restrict__ w2,
                             const float* __restrict__ b2, float* out, int E) {
    int tid  = threadIdx.x;
    int c    = tid & 15;                              // chunk within row
    int e    = blockIdx.x * (blockDim.x >> 4) + (tid >> 4);
    float acc = 0.0f;
    bool valid = (e < E);
    if (valid) {
        int s = ei[e];
        int d = ei[E + e];
        float4 w = ((const float4*)w2)[c];
        float4 a = ((const float4*)(u + (size_t)s * HID))[c];
        float4 b = ((const float4*)(v + (size_t)d * HID))[c];
        acc = fmaxf(a.x + b.x, 0.0f) * w.x
            + fmaxf(a.y + b.y, 0.0f) * w.y
            + fmaxf(a.z + b.z, 0.0f) * w.z
            + fmaxf(a.w + b.w, 0.0f) * w.w;
    }
#pragma unroll
    for (int off = 8; off >= 1; off >>= 1)            // stays within each 16-lane group
        acc += __shfl_xor(acc, off, 32);
    if (valid && c == 0) out[e] = acc + b2[0];
}

// ---------------- launcher ----------------

static inline size_t alignUp(size_t x, size_t a) { return (x + a - 1) & ~(a - 1); }

extern "C" void kernel_launch(void* const* d_in, const int* in_sizes, int n_in,
                              void* d_out, int out_size, void* d_ws, size_t ws_size,
                              hipStream_t stream) {
    const float* x    = (const float*)d_in[0];
    const int*   ei   = (const int*)d_in[1];     // (2,E) int32
    const float* W1   = (const float*)d_in[2];
    const float* b1   = (const float*)d_in[3];
    const float* W2   = (const float*)d_in[4];
    const float* b2   = (const float*)d_in[5];
    const float* lpW1 = (const float*)d_in[6];   // (128,64)
    const float* lpb1 = (const float*)d_in[7];
    const float* lpW2 = (const float*)d_in[8];   // (64,1)
    const float* lpb2 = (const float*)d_in[9];
    float* out = (float*)d_out;

    const int N = in_sizes[0] / 4;               // IN_CH = 4
    const int E = in_sizes[1] / 2;

    // workspace carve-up (all 256B aligned)
    char* ws = (char*)d_ws;
    size_t off = 0;
    int*   deg  = (int*)(ws + off);   off = alignUp(off + (size_t)N * 4, 256);
    float* dinv = (float*)(ws + off); off = alignUp(off + (size_t)N * 4, 256);
    float* nrm  = (float*)(ws + off); off = alignUp(off + (size_t)E * 4, 256);
    float* bufA = (float*)(ws + off); off = alignUp(off + (size_t)N * HID * 4, 256);
    float* bufB = (float*)(ws + off); off = alignUp(off + (size_t)N * HID * 4, 256);
    float* bufC = (float*)(ws + off); off = alignUp(off + (size_t)N * HID * 4, 256);
    (void)ws_size; (void)n_in; (void)out_size;

    const int TB = 256;
    dim3 gB(TB);
    dim3 gemmBlk(32, 8);
    int slabs = N / 16;                           // N is a multiple of 16
    dim3 gemmGrid((slabs + 7) / 8);
    int nc = N * HID;

    // degrees / normalization
    k_init_deg <<<(N + TB - 1) / TB, gB, 0, stream>>>(deg, N);
    k_count_deg<<<(E + TB - 1) / TB, gB, 0, stream>>>(ei, deg, E);
    k_dinv     <<<(N + TB - 1) / TB, gB, 0, stream>>>(deg, dinv, N);
    k_norm     <<<(E + TB - 1) / TB, gB, 0, stream>>>(ei, dinv, nrm, E);

    // layer 1: hw1 = x @ W1  (K=4)
    k_gemm64<<<gemmGrid, gemmBlk, 0, stream>>>(x, W1, nullptr, bufA, N, 4, 0);
    k_selfinit <<<(nc + TB - 1) / TB, gB, 0, stream>>>(bufA, dinv, bufB, N);
    k_scatter  <<<(E * 16 + TB - 1) / TB, gB, 0, stream>>>(ei, nrm, bufA, bufB, E);
    k_bias_relu<<<(nc + TB - 1) / TB, gB, 0, stream>>>(bufB, b1, N);      // h1 in bufB

    // layer 2: hw2 = h1 @ W2  (K=64)
    k_gemm64<<<gemmGrid, gemmBlk, 0, stream>>>(bufB, W2, nullptr, bufC, N, HID, 0);
    k_selfinit <<<(nc + TB - 1) / TB, gB, 0, stream>>>(bufC, dinv, bufA, N);
    k_scatter  <<<(E * 16 + TB - 1) / TB, gB, 0, stream>>>(ei, nrm, bufC, bufA, E);
    k_bias_relu<<<(nc + TB - 1) / TB, gB, 0, stream>>>(bufA, b2, N);      // h2 in bufA

    // edge-MLP factorization: u = h2 @ lpW1[0:64] + lpb1 ; v = h2 @ lpW1[64:128]
    k_gemm64<<<gemmGrid, gemmBlk, 0, stream>>>(bufA, lpW1,            lpb1,    bufB, N, HID, 0);
    k_gemm64<<<gemmGrid, gemmBlk, 0, stream>>>(bufA, lpW1 + 64 * HID, nullptr, bufC, N, HID, 0);

    // per-edge scoring: 16 lanes/edge -> blocks of 256 threads cover 16 edges
    k_edge_score<<<(E + 15) / 16, gB, 0, stream>>>(ei, bufB, bufC, lpW2, lpb2, out, E);
}